// WeightedProcrustes_2173253452365
// MI455X (gfx1250) — compile-verified
//
#include <hip/hip_runtime.h>

// Weighted Procrustes, B=8192 batches of N=512 3D points.
// Bandwidth-bound (~117MB in -> ~5us floor @ 23.3TB/s).
// Strategy: per wave, 4 batches. All 16 weighted moments per batch
// (S, Sw*src, Sw*tgt, Sw*src*tgt^T) = homogeneous moment matrix
// G = sum_n w_n * u_n v_n^T, u=(s,1), v=(t,1), computed as a chain of
// V_WMMA_F32_16X16X4_F32 rank-4 updates; batch b lives in diagonal
// block rows/cols [4b,4b+3] of the 16x16 accumulator. Then lanes 0-3
// do the tiny 3x3 Jacobi SVD + Kabsch assembly in registers.

typedef __attribute__((ext_vector_type(2))) float v2f;
typedef __attribute__((ext_vector_type(8))) float v8f;

#define NBATCH 8192
#define NPTS   512
#define EPSW   1e-5f

__device__ __forceinline__ v8f moment_step(const float* __restrict__ wb,
                                           const float* __restrict__ sb,
                                           const float* __restrict__ tb,
                                           int p, int isOne, v8f acc) {
  // p = first of this lane's two K-slots (K=0,1 for lanes 0-15; K=2,3 for 16-31)
  float w0 = wb[p];
  float w1 = wb[p + 1];
  w0 = (w0 < 0.f) ? 0.f : w0;           // WEIGHT_THRESHOLD = 0
  w1 = (w1 < 0.f) ? 0.f : w1;
  float s0 = sb[p * 3];
  float s1 = sb[(p + 1) * 3];
  float t0 = tb[p * 3];
  float t1 = tb[(p + 1) * 3];
  v2f a, b;
  a.x = isOne ? w0 : w0 * s0;           // A = w * u_comp   (u = (src,1))
  a.y = isOne ? w1 : w1 * s1;
  b.x = isOne ? 1.f : t0;               // B = v_comp       (v = (tgt,1))
  b.y = isOne ? 1.f : t1;
  return __builtin_amdgcn_wmma_f32_16x16x4_f32(false, a, false, b,
                                               (short)0, acc, false, false);
}

// one Jacobi rotation on (p,q), r = remaining index; all names are lvalues
#define JROT(kpp, kpq, kqq, kpr, kqr, vap, vaq, vbp, vbq, vcp, vcq)      \
  {                                                                      \
    float apq_ = kpq;                                                    \
    if (fabsf(apq_) > 1e-20f) {                                          \
      float app_ = kpp, aqq_ = kqq;                                      \
      float tau_ = (aqq_ - app_) / (2.f * apq_);                         \
      float rt_ = sqrtf(1.f + tau_ * tau_);                              \
      float tj_ = (tau_ >= 0.f) ? 1.f / (tau_ + rt_)                     \
                                : 1.f / (tau_ - rt_);                    \
      float cj_ = 1.f / sqrtf(1.f + tj_ * tj_);                          \
      float sj_ = tj_ * cj_;                                             \
      kpp = app_ - tj_ * apq_;                                           \
      kqq = aqq_ + tj_ * apq_;                                           \
      kpq = 0.f;                                                         \
      float x0_ = kpr, x1_ = kqr;                                        \
      kpr = cj_ * x0_ - sj_ * x1_;                                       \
      kqr = sj_ * x0_ + cj_ * x1_;                                       \
      float tv_;                                                         \
      tv_ = vap; vap = cj_ * tv_ - sj_ * vaq; vaq = sj_ * tv_ + cj_ * vaq; \
      tv_ = vbp; vbp = cj_ * tv_ - sj_ * vbq; vbq = sj_ * tv_ + cj_ * vbq; \
      tv_ = vcp; vcp = cj_ * tv_ - sj_ * vcq; vcq = sj_ * tv_ + cj_ * vcq; \
    }                                                                    \
  }

#define CSWAP3(la, lb, ax, ay, az, bx, by, bz)                           \
  if (la < lb) {                                                         \
    float t_;                                                            \
    t_ = la; la = lb; lb = t_;                                           \
    t_ = ax; ax = bx; bx = t_;                                           \
    t_ = ay; ay = by; by = t_;                                           \
    t_ = az; az = bz; bz = t_;                                           \
  }

extern "C" __global__ __launch_bounds__(32)
void wproc_kernel(const float* __restrict__ src,
                  const float* __restrict__ tgt,
                  const float* __restrict__ wgt,
                  float* __restrict__ out) {
  __shared__ float tile[256];

  const int lane = threadIdx.x;         // 0..31
  const int idx  = lane & 15;           // = M (A) = N (B) position
  const int binw = idx >> 2;            // batch within wave, 0..3
  const int comp = idx & 3;             // 0..2 = xyz, 3 = homogeneous 1
  const int poff = (lane & 16) ? 2 : 0; // K-slot base for this half-wave
  const int b    = blockIdx.x * 4 + binw;

  const int cc = (comp < 3) ? comp : 0; // safe address for comp==3 lanes
  const float* wb = wgt + b * NPTS;
  const float* sb = src + b * NPTS * 3 + cc;
  const float* tb = tgt + b * NPTS * 3 + cc;
  const int isOne = (comp == 3);

  v8f acc0 = {}; v8f acc1 = {}; v8f acc2 = {}; v8f acc3 = {};
  for (int n = 0; n < NPTS; n += 16) {
    acc0 = moment_step(wb, sb, tb, n +  0 + poff, isOne, acc0);
    acc1 = moment_step(wb, sb, tb, n +  4 + poff, isOne, acc1);
    acc2 = moment_step(wb, sb, tb, n +  8 + poff, isOne, acc2);
    acc3 = moment_step(wb, sb, tb, n + 12 + poff, isOne, acc3);
  }
  v8f acc = (acc0 + acc1) + (acc2 + acc3);

  // Unstripe D tile to LDS: lane L, VGPR r -> D[row=(L<16?r:r+8)][col=L&15]
  const int rowbase = (lane < 16) ? 0 : 8;
#pragma unroll
  for (int r = 0; r < 8; ++r)
    tile[(rowbase + r) * 16 + idx] = acc[r];

  __syncthreads();

  if (lane < 4) {
    const int bb = blockIdx.x * 4 + lane;
    const int o  = lane * 4;
    float G[4][4];
#pragma unroll
    for (int i = 0; i < 4; ++i)
#pragma unroll
      for (int j = 0; j < 4; ++j)
        G[i][j] = tile[(o + i) * 16 + (o + j)];

    // G = [[M9, sx],[tx^T, S]]; wn = w/(S+eps)
    const float S   = G[3][3];
    const float inv = 1.f / (S + EPSW);
    const float scx = G[0][3] * inv, scy = G[1][3] * inv, scz = G[2][3] * inv;
    const float tcx = G[3][0] * inv, tcy = G[3][1] * inv, tcz = G[3][2] * inv;
    const float f   = 2.f - S * inv;  // H = M9*inv - (2-s) * src_c tgt_c^T
    float h00 = G[0][0] * inv - f * scx * tcx;
    float h01 = G[0][1] * inv - f * scx * tcy;
    float h02 = G[0][2] * inv - f * scx * tcz;
    float h10 = G[1][0] * inv - f * scy * tcx;
    float h11 = G[1][1] * inv - f * scy * tcy;
    float h12 = G[1][2] * inv - f * scy * tcz;
    float h20 = G[2][0] * inv - f * scz * tcx;
    float h21 = G[2][1] * inv - f * scz * tcy;
    float h22 = G[2][2] * inv - f * scz * tcz;

    // K = H^T H (symmetric PSD)
    float k00 = h00 * h00 + h10 * h10 + h20 * h20;
    float k01 = h00 * h01 + h10 * h11 + h20 * h21;
    float k02 = h00 * h02 + h10 * h12 + h20 * h22;
    float k11 = h01 * h01 + h11 * h11 + h21 * h21;
    float k12 = h01 * h02 + h11 * h12 + h21 * h22;
    float k22 = h02 * h02 + h12 * h12 + h22 * h22;

    // V = I; columns are eigenvectors
    float v00 = 1.f, v01 = 0.f, v02 = 0.f;
    float v10 = 0.f, v11 = 1.f, v12 = 0.f;
    float v20 = 0.f, v21 = 0.f, v22 = 1.f;

#pragma unroll
    for (int sw = 0; sw < 6; ++sw) {
      JROT(k00, k01, k11, k02, k12, v00, v01, v10, v11, v20, v21); // (0,1)
      JROT(k00, k02, k22, k01, k12, v00, v02, v10, v12, v20, v22); // (0,2)
      JROT(k11, k12, k22, k01, k02, v01, v02, v11, v12, v21, v22); // (1,2)
    }

    // sort eigenpairs descending
    float l0 = k00, l1 = k11, l2 = k22;
    float c0x = v00, c0y = v10, c0z = v20;
    float c1x = v01, c1y = v11, c1z = v21;
    float c2x = v02, c2y = v12, c2z = v22;
    CSWAP3(l0, l1, c0x, c0y, c0z, c1x, c1y, c1z);
    CSWAP3(l1, l2, c1x, c1y, c1z, c2x, c2y, c2z);
    CSWAP3(l0, l1, c0x, c0y, c0z, c1x, c1y, c1z);

    // U columns: u_i = H v_i / sigma_i (normalize + Gram-Schmidt);
    // third columns via cross products => proper rotation (== diag(1,1,det) fix)
    float u0x = h00 * c0x + h01 * c0y + h02 * c0z;
    float u0y = h10 * c0x + h11 * c0y + h12 * c0z;
    float u0z = h20 * c0x + h21 * c0y + h22 * c0z;
    float rn0 = rsqrtf(fmaxf(u0x * u0x + u0y * u0y + u0z * u0z, 1e-30f));
    u0x *= rn0; u0y *= rn0; u0z *= rn0;

    float u1x = h00 * c1x + h01 * c1y + h02 * c1z;
    float u1y = h10 * c1x + h11 * c1y + h12 * c1z;
    float u1z = h20 * c1x + h21 * c1y + h22 * c1z;
    float pr  = u1x * u0x + u1y * u0y + u1z * u0z;
    u1x -= pr * u0x; u1y -= pr * u0y; u1z -= pr * u0z;
    float rn1 = rsqrtf(fmaxf(u1x * u1x + u1y * u1y + u1z * u1z, 1e-30f));
    u1x *= rn1; u1y *= rn1; u1z *= rn1;

    float u2x = u0y * u1z - u0z * u1y;
    float u2y = u0z * u1x - u0x * u1z;
    float u2z = u0x * u1y - u0y * u1x;

    float c2xx = c0y * c1z - c0z * c1y;
    float c2yy = c0z * c1x - c0x * c1z;
    float c2zz = c0x * c1y - c0y * c1x;

    // R = v0 u0^T + v1 u1^T + v2 u2^T
    float r00 = c0x * u0x + c1x * u1x + c2xx * u2x;
    float r01 = c0x * u0y + c1x * u1y + c2xx * u2y;
    float r02 = c0x * u0z + c1x * u1z + c2xx * u2z;
    float r10 = c0y * u0x + c1y * u1x + c2yy * u2x;
    float r11 = c0y * u0y + c1y * u1y + c2yy * u2y;
    float r12 = c0y * u0z + c1y * u1z + c2yy * u2z;
    float r20 = c0z * u0x + c1z * u1x + c2zz * u2x;
    float r21 = c0z * u0y + c1z * u1y + c2zz * u2y;
    float r22 = c0z * u0z + c1z * u1z + c2zz * u2z;

    // t = tgt_c - R src_c
    float txo = tcx - (r00 * scx + r01 * scy + r02 * scz);
    float tyo = tcy - (r10 * scx + r11 * scy + r12 * scz);
    float tzo = tcz - (r20 * scx + r21 * scy + r22 * scz);

    float* Ro = out + bb * 9;
    Ro[0] = r00; Ro[1] = r01; Ro[2] = r02;
    Ro[3] = r10; Ro[4] = r11; Ro[5] = r12;
    Ro[6] = r20; Ro[7] = r21; Ro[8] = r22;
    float* To = out + NBATCH * 9 + bb * 3;
    To[0] = txo; To[1] = tyo; To[2] = tzo;
  }
}

extern "C" void kernel_launch(void* const* d_in, const int* in_sizes, int n_in,
                              void* d_out, int out_size, void* d_ws, size_t ws_size,
                              hipStream_t stream) {
  (void)in_sizes; (void)n_in; (void)out_size; (void)d_ws; (void)ws_size;
  const float* src = (const float*)d_in[0];
  const float* tgt = (const float*)d_in[1];
  const float* wgt = (const float*)d_in[2];
  float* out = (float*)d_out;
  hipLaunchKernelGGL(wproc_kernel, dim3(NBATCH / 4), dim3(32), 0, stream,
                     src, tgt, wgt, out);
}